// GCNTrialClassifier_68874095558731
// MI455X (gfx1250) — compile-verified
//
#include <hip/hip_runtime.h>
#include <hip/hip_bf16.h>

typedef __attribute__((ext_vector_type(2))) float v2f;
typedef __attribute__((ext_vector_type(8))) float v8f;

#define N_NODES  100000
#define N_EDGES  1600000
#define N_GRAPHS 1000
#define IN_DIM   128
#define HID      100
#define HPAD     112   // 7 WMMA tiles of 16

// ---------------- degree / normalization ----------------
__global__ void k_init_deg(float* deg, int n) {
    int i = blockIdx.x * blockDim.x + threadIdx.x;
    if (i < n) deg[i] = 1.0f;                      // self-loop contribution
}

__global__ void k_edge_deg(const int* __restrict__ dst, float* deg, int e) {
    int i = blockIdx.x * blockDim.x + threadIdx.x;
    if (i < e) atomicAdd(&deg[dst[i]], 1.0f);
}

__global__ void k_dinv(float* deg, int n) {
    int i = blockIdx.x * blockDim.x + threadIdx.x;
    if (i < n) deg[i] = rsqrtf(deg[i]);            // deg >= 1 always
}

// ---------------- weight packing ----------------
// Pack W[K x sc] -> Wp[(K/2) x HPAD] of float2: Wp[p][n] = {W[2p][n], W[2p+1][n]},
// zero-padded so a B fragment is ONE aligned ds_load_b64 in the GEMM.
__global__ void k_pack_w(const float* __restrict__ src, float2* __restrict__ dst,
                         int sr, int sc, int kpairs) {
    int i = blockIdx.x * blockDim.x + threadIdx.x;
    if (i >= kpairs * HPAD) return;
    int p = i / HPAD, n = i % HPAD;
    int k0 = 2 * p, k1 = 2 * p + 1;
    float lo = (k0 < sr && n < sc) ? src[k0 * sc + n] : 0.0f;
    float hi = (k1 < sr && n < sc) ? src[k1 * sc + n] : 0.0f;
    dst[i] = make_float2(lo, hi);
}

__global__ void k_pad_vec(const float* __restrict__ src, float* __restrict__ dst,
                          int sc, int dc) {
    int i = blockIdx.x * blockDim.x + threadIdx.x;
    if (i < dc) dst[i] = (i < sc) ? src[i] : 0.0f;
}

__global__ void k_zero(float* p, size_t n) {
    size_t i = (size_t)blockIdx.x * blockDim.x + threadIdx.x;
    size_t stride = (size_t)gridDim.x * blockDim.x;
    for (; i < n; i += stride) p[i] = 0.0f;
}

// ---------------- fp32 WMMA GEMM: H[nrows x HPAD] = X[nrows x K] @ W[K x HPAD] ----------------
// 8 wave32 per block; each wave owns a 16-row M tile, 7 N tiles of 16.
// Wp is the K-pair-interleaved float2 packing produced by k_pack_w.
__global__ __launch_bounds__(256) void k_gemm_wmma(
    const float* __restrict__ X, const float2* __restrict__ Wp,
    float* __restrict__ H, int nrows, int K)
{
    __shared__ float2 wl[(IN_DIM / 2) * HPAD];     // up to 57,344 B of the 320 KB WGP LDS
    int tid = threadIdx.x;
    int kpairs = K >> 1;
    for (int i = tid; i < kpairs * HPAD; i += 256) wl[i] = Wp[i];
    __syncthreads();

    int lane = tid & 31;
    int wave = tid >> 5;
    int m    = lane & 15;
    int koff = (lane >> 4) << 1;                   // lanes 0-15: K+0/K+1, lanes 16-31: K+2/K+3
    int row0 = blockIdx.x * 128 + wave * 16;
    int row  = row0 + m;
    bool inr = row < nrows;
    int rowc = inr ? row : (nrows - 1);            // clamp: keep loads valid, no exec branch
    const float* xrow = X + (size_t)rowc * K;

    v8f acc[7];
#pragma unroll
    for (int t = 0; t < 7; ++t)
        acc[t] = (v8f){0.f, 0.f, 0.f, 0.f, 0.f, 0.f, 0.f, 0.f};

    for (int k = 0; k < K; k += 4) {
        float2 t2 = *(const float2*)(xrow + k + koff);
        v2f a = (v2f){ inr ? t2.x : 0.f, inr ? t2.y : 0.f };   // v_cndmask, no branch
        const float2* wrow = wl + ((k + koff) >> 1) * HPAD;    // K-pair row
#pragma unroll
        for (int t = 0; t < 7; ++t) {
            float2 bb = wrow[t * 16 + m];                      // one ds_load_b64
            v2f b = (v2f){bb.x, bb.y};
            // D = A(16x4,f32) * B(4x16,f32) + C(16x16,f32)
            acc[t] = __builtin_amdgcn_wmma_f32_16x16x4_f32(
                false, a, false, b, (short)0, acc[t], false, false);
        }
    }

    // C/D layout: VGPR v, lanes 0-15 -> M=v; lanes 16-31 -> M=v+8; N = lane&15
    int rbase = row0 + ((lane >> 4) << 3);
#pragma unroll
    for (int t = 0; t < 7; ++t) {
        int cc = t * 16 + m;
#pragma unroll
        for (int v = 0; v < 8; ++v) {
            int rr = rbase + v;
            if (rr < nrows) H[(size_t)rr * HPAD + cc] = acc[t][v];
        }
    }
}

// ---------------- gather-normalize-scatter (one wave32 per edge/self-loop) ----------------
__global__ __launch_bounds__(256) void k_scatter(
    const float* __restrict__ H, const float* __restrict__ dinv,
    const int* __restrict__ src, const int* __restrict__ dst,
    float* __restrict__ AGG, int nE, int nN)
{
    long long gid = (long long)blockIdx.x * 256 + threadIdx.x;
    int e    = (int)(gid >> 5);
    int lane = (int)(gid & 31);
    if (e >= nE + nN) return;
    int s, d;
    if (e < nE) { s = src[e]; d = dst[e]; }
    else        { s = d = e - nE; }                 // self-loop
    float w = dinv[s] * dinv[d];
    const float* hs = H + (size_t)s * HPAD;
    float*       ad = AGG + (size_t)d * HPAD;
    atomicAdd(&ad[lane],      hs[lane]      * w);
    atomicAdd(&ad[lane + 32], hs[lane + 32] * w);
    atomicAdd(&ad[lane + 64], hs[lane + 64] * w);
    if (lane < 16) atomicAdd(&ad[lane + 96], hs[lane + 96] * w);
}

__global__ void k_bias_relu(float* __restrict__ A, const float* __restrict__ b,
                            long long n) {
    long long i = (long long)blockIdx.x * blockDim.x + threadIdx.x;
    if (i >= n) return;
    int c = (int)(i % HPAD);
    float v = A[i] + b[c];
    A[i] = v > 0.f ? v : 0.f;
}

// ---------------- global mean pool ----------------
__global__ void k_counts(const int* __restrict__ batch, float* counts, int n) {
    int i = blockIdx.x * blockDim.x + threadIdx.x;
    if (i < n) atomicAdd(&counts[batch[i]], 1.0f);
}

__global__ void k_pool_sum(const float* __restrict__ X, const int* __restrict__ batch,
                           float* __restrict__ pool, int n) {
    long long i = (long long)blockIdx.x * blockDim.x + threadIdx.x;
    if (i >= (long long)n * HID) return;
    int node = (int)(i / HID);
    int c    = (int)(i % HID);
    atomicAdd(&pool[(size_t)batch[node] * HID + c], X[(size_t)node * HPAD + c]);
}

__global__ void k_pool_fin(const float* __restrict__ pool, const float* __restrict__ counts,
                           float* __restrict__ out, int n) {
    int i = blockIdx.x * blockDim.x + threadIdx.x;
    if (i >= n) return;
    int g = i / HID;
    out[i] = pool[i] / fmaxf(counts[g], 1.0f);
}

// ---------------- launcher ----------------
extern "C" void kernel_launch(void* const* d_in, const int* in_sizes, int n_in,
                              void* d_out, int out_size, void* d_ws, size_t ws_size,
                              hipStream_t stream) {
    const float* x  = (const float*)d_in[0];    // [N, 128]
    const float* W1 = (const float*)d_in[1];    // [128, 100]
    const float* b1 = (const float*)d_in[2];    // [100]
    const float* W2 = (const float*)d_in[3];    // [100, 100]
    const float* b2 = (const float*)d_in[4];    // [100]
    const int*   ei = (const int*)d_in[5];      // [2, E] flat: src then dst
    const int*   bt = (const int*)d_in[6];      // [N]
    const int* esrc = ei;
    const int* edst = ei + N_EDGES;
    float* out = (float*)d_out;

    // workspace carve-up (floats; float2 regions are 8-byte aligned by construction)
    float*  ws   = (float*)d_ws;
    float*  dinv = ws;                               // N
    float2* W1p  = (float2*)(dinv + N_NODES);        // (128/2)*112 float2
    float*  b1p  = (float*)(W1p + (IN_DIM / 2) * HPAD);  // 112
    float2* W2p  = (float2*)(b1p + HPAD);            // (112/2)*112 float2
    float*  b2p  = (float*)(W2p + (HPAD / 2) * HPAD);    // 112
    float*  B0   = b2p + HPAD;                       // N*112  (H1, then AGG2)
    float*  B1   = B0 + (size_t)N_NODES * HPAD;      // N*112  (AGG1 -> X2)
    float*  B2   = B1 + (size_t)N_NODES * HPAD;      // N*112  (H2)
    float*  pool = B2 + (size_t)N_NODES * HPAD;      // G*100
    float*  cnts = pool + N_GRAPHS * HID;            // G

    const size_t NH = (size_t)N_NODES * HPAD;
    auto cdiv = [](long long a, long long b) { return (unsigned)((a + b - 1) / b); };

    // 1) symmetric normalization coefficients
    k_init_deg<<<cdiv(N_NODES, 256), 256, 0, stream>>>(dinv, N_NODES);
    k_edge_deg<<<cdiv(N_EDGES, 256), 256, 0, stream>>>(edst, dinv, N_EDGES);
    k_dinv<<<cdiv(N_NODES, 256), 256, 0, stream>>>(dinv, N_NODES);

    // 2) pack weights into K-pair-interleaved float2, pad biases
    k_pack_w<<<cdiv((IN_DIM / 2) * HPAD, 256), 256, 0, stream>>>(W1, W1p, IN_DIM, HID, IN_DIM / 2);
    k_pad_vec<<<1, HPAD, 0, stream>>>(b1, b1p, HID, HPAD);
    k_pack_w<<<cdiv((HPAD / 2) * HPAD, 256), 256, 0, stream>>>(W2, W2p, HID, HID, HPAD / 2);
    k_pad_vec<<<1, HPAD, 0, stream>>>(b2, b2p, HID, HPAD);

    // 3) layer 1: H1 = X @ W1  (fp32 WMMA), aggregate, bias+relu
    k_gemm_wmma<<<cdiv(N_NODES, 128), 256, 0, stream>>>(x, W1p, B0, N_NODES, IN_DIM);
    k_zero<<<4096, 256, 0, stream>>>(B1, NH);
    k_scatter<<<cdiv((long long)(N_EDGES + N_NODES) * 32, 256), 256, 0, stream>>>(
        B0, dinv, esrc, edst, B1, N_EDGES, N_NODES);
    k_bias_relu<<<cdiv((long long)NH, 256), 256, 0, stream>>>(B1, b1p, (long long)NH);

    // 4) layer 2: H2 = X2 @ W2 (K = HPAD since padded cols are zero), aggregate, bias+relu
    k_gemm_wmma<<<cdiv(N_NODES, 128), 256, 0, stream>>>(B1, W2p, B2, N_NODES, HPAD);
    k_zero<<<4096, 256, 0, stream>>>(B0, NH);
    k_scatter<<<cdiv((long long)(N_EDGES + N_NODES) * 32, 256), 256, 0, stream>>>(
        B2, dinv, esrc, edst, B0, N_EDGES, N_NODES);
    k_bias_relu<<<cdiv((long long)NH, 256), 256, 0, stream>>>(B0, b2p, (long long)NH);

    // 5) global mean pool -> d_out [G, 100] (== reshape(-1,25,4) flat)
    k_zero<<<64, 256, 0, stream>>>(pool, (size_t)N_GRAPHS * HID + N_GRAPHS);
    k_counts<<<cdiv(N_NODES, 256), 256, 0, stream>>>(bt, cnts, N_NODES);
    k_pool_sum<<<cdiv((long long)N_NODES * HID, 256), 256, 0, stream>>>(B0, bt, pool, N_NODES);
    k_pool_fin<<<cdiv(N_GRAPHS * HID, 256), 256, 0, stream>>>(pool, cnts, out, N_GRAPHS * HID);
}